// ProteinCNN_73254962200756
// MI455X (gfx1250) — compile-verified
//
#include <hip/hip_runtime.h>

#define HDIM 128
#define GNUM 128
#define CLSD 128
#define OUTD 64
#define EPSV 1e-5f

typedef __attribute__((ext_vector_type(2))) float v2f;
typedef __attribute__((ext_vector_type(8))) float v8f;

// ---------------------------------------------------------------------------
// WMMA f32 GEMM: C[M,NC] = A[M,K] @ B[K,NC] (+bias) (+leaky)
// RT = row tiles per wave. grid = (M/(16*RT), ceil(NC/64)), block = 128 (4 waves).
// Each wave computes (16*RT) x 16 output; B operand loaded once per k-step and
// reused across RT WMMAs (loads/WMMA = 2/RT + 2), RT accumulator chains hide
// the XDL D->A latency.
// A-matrix 16x4 f32 layout: lanes 0-15 -> K{0,1}, lanes 16-31 -> K{2,3}
// C/D 16x16 f32 layout: VGPR r -> row r (lanes 0-15) / row r+8 (lanes 16-31)
// ---------------------------------------------------------------------------
template <int RT>
__global__ __launch_bounds__(128) void k_gemm_wmma(
    const float* __restrict__ A, const float* __restrict__ B,
    const float* __restrict__ bias, float* __restrict__ C,
    int M, int K, int NC, int act) {
  const int lane = threadIdx.x & 31;
  const int wave = threadIdx.x >> 5;
  const int rowBase = blockIdx.x * (16 * RT);
  const int colBase = blockIdx.y * 64 + wave * 16;
  if (colBase >= NC) return;  // wave-uniform

  const int mrow = rowBase + (lane & 15);
  const int col  = colBase + (lane & 15);
  const int koff = (lane >> 4) * 2;  // lanes 16-31 hold K+2,K+3

  v8f acc[RT];
#pragma unroll
  for (int t = 0; t < RT; ++t) acc[t] = (v8f){};

  for (int k0 = 0; k0 < K; k0 += 4) {
    v2f b;
    b.x = B[(size_t)(k0 + koff) * NC + col];
    b.y = B[(size_t)(k0 + koff + 1) * NC + col];
#pragma unroll
    for (int t = 0; t < RT; ++t) {
      const float* arow = A + (size_t)(mrow + 16 * t) * K;
      v2f a;
      a.x = arow[k0 + koff];
      a.y = arow[k0 + koff + 1];
      acc[t] = __builtin_amdgcn_wmma_f32_16x16x4_f32(
          /*neg_a=*/false, a, /*neg_b=*/false, b,
          /*c_mod=*/(short)0, acc[t], /*reuse_a=*/false, /*reuse_b=*/false);
    }
  }

  const int rbase = (lane >> 4) * 8;
  float bcol = bias ? bias[col] : 0.0f;
#pragma unroll
  for (int t = 0; t < RT; ++t) {
#pragma unroll
    for (int r = 0; r < 8; ++r) {
      int row = rowBase + 16 * t + rbase + r;
      float v = acc[t][r] + bcol;
      if (act == 1) v = (v >= 0.0f) ? v : 0.01f * v;
      C[(size_t)row * NC + col] = v;
    }
  }
}

// -------------------- degree / normalization --------------------
__global__ void k_deg(const int* __restrict__ dst, float* __restrict__ deg, int E) {
  for (int i = blockIdx.x * blockDim.x + threadIdx.x; i < E;
       i += gridDim.x * blockDim.x)
    atomicAdd(&deg[dst[i]], 1.0f);
}

__global__ void k_dis(float* __restrict__ deg, int n) {
  int i = blockIdx.x * blockDim.x + threadIdx.x;
  if (i < n) deg[i] = rsqrtf(deg[i] + 1.0f);  // +1 self-loop; deg>=1 so no max
}

// ------------- edge scatter: agg[dst] += h[src]*norm (1 wave / edge) ----------
// lane -> float4 of the 128-wide feature row: one b128 load + 4 f32 atomics
__global__ __launch_bounds__(256) void k_scatter(
    const float4* __restrict__ h4, const int* __restrict__ src,
    const int* __restrict__ dst, const float* __restrict__ dis,
    float* __restrict__ agg, int E) {
  int e = blockIdx.x * 8 + (threadIdx.x >> 5);
  if (e >= E) return;
  int lane = threadIdx.x & 31;
  int s = src[e], d = dst[e];
  float norm = dis[s] * dis[d];
  float4 v = h4[(size_t)s * (HDIM / 4) + lane];
  float* ap = agg + (size_t)d * HDIM + lane * 4;
  atomicAdd(ap + 0, v.x * norm);
  atomicAdd(ap + 1, v.y * norm);
  atomicAdd(ap + 2, v.z * norm);
  atomicAdd(ap + 3, v.w * norm);
}

// ------ fused: self-loop + bias, write back, accumulate BN column stats -------
// block = 128 threads = 4 row-groups x 32 lanes; lane handles 4 columns (b128)
__global__ __launch_bounds__(128) void k_finalize(
    float4* __restrict__ agg4, const float4* __restrict__ h4,
    const float* __restrict__ dis, const float* __restrict__ bias,
    float* __restrict__ colsum, float* __restrict__ colsq, int n) {
  const int lane = threadIdx.x & 31;
  const int grp = threadIdx.x >> 5;
  const int c4 = lane * 4;
  const float4 bc = *(const float4*)(bias + c4);
  float s0 = 0, s1 = 0, s2 = 0, s3 = 0;
  float q0 = 0, q1 = 0, q2 = 0, q3 = 0;
  for (int i = blockIdx.x * 4 + grp; i < n; i += gridDim.x * 4) {
    float di = dis[i];
    float dd = di * di;
    size_t idx = (size_t)i * (HDIM / 4) + lane;
    float4 a = agg4[idx];
    float4 hv = h4[idx];
    float y0 = a.x + hv.x * dd + bc.x;
    float y1 = a.y + hv.y * dd + bc.y;
    float y2 = a.z + hv.z * dd + bc.z;
    float y3 = a.w + hv.w * dd + bc.w;
    agg4[idx] = make_float4(y0, y1, y2, y3);
    s0 += y0; q0 += y0 * y0;
    s1 += y1; q1 += y1 * y1;
    s2 += y2; q2 += y2 * y2;
    s3 += y3; q3 += y3 * y3;
  }
  atomicAdd(&colsum[c4 + 0], s0); atomicAdd(&colsq[c4 + 0], q0);
  atomicAdd(&colsum[c4 + 1], s1); atomicAdd(&colsq[c4 + 1], q1);
  atomicAdd(&colsum[c4 + 2], s2); atomicAdd(&colsq[c4 + 2], q2);
  atomicAdd(&colsum[c4 + 3], s3); atomicAdd(&colsq[c4 + 3], q3);
}

__global__ void k_bnstats(const float* __restrict__ colsum,
                          const float* __restrict__ colsq,
                          const float* __restrict__ g, const float* __restrict__ b,
                          float* __restrict__ scale, float* __restrict__ shift,
                          int n) {
  int c = threadIdx.x;
  float inv_n = 1.0f / (float)n;
  float mean = colsum[c] * inv_n;
  float var = colsq[c] * inv_n - mean * mean;
  float sc = g[c] * rsqrtf(var + EPSV);
  scale[c] = sc;
  shift[c] = b[c] - mean * sc;
}

__global__ void k_bnrelu(float4* __restrict__ x, const float* __restrict__ scale,
                         const float* __restrict__ shift, int total4) {
  for (int i = blockIdx.x * blockDim.x + threadIdx.x; i < total4;
       i += gridDim.x * blockDim.x) {
    int c = (i * 4) & (HDIM - 1);
    float4 v = x[i];
    v.x = fmaxf(v.x * scale[c + 0] + shift[c + 0], 0.0f);
    v.y = fmaxf(v.y * scale[c + 1] + shift[c + 1], 0.0f);
    v.z = fmaxf(v.z * scale[c + 2] + shift[c + 2], 0.0f);
    v.w = fmaxf(v.w * scale[c + 3] + shift[c + 3], 0.0f);
    x[i] = v;
  }
}

// -------------------- global mean pool --------------------
__global__ __launch_bounds__(HDIM) void k_pool(
    const float* __restrict__ x, const int* __restrict__ batch,
    float* __restrict__ pooled, float* __restrict__ cnt, int n) {
  int c = threadIdx.x;
  for (int i = blockIdx.x; i < n; i += gridDim.x) {
    int g = batch[i];
    atomicAdd(&pooled[(size_t)g * HDIM + c], x[(size_t)i * HDIM + c]);
    if (c == 0) atomicAdd(&cnt[g], 1.0f);
  }
}

__global__ void k_pooldiv(float* __restrict__ pooled, const float* __restrict__ cnt) {
  int g = blockIdx.x, c = threadIdx.x;
  pooled[(size_t)g * HDIM + c] /= fmaxf(cnt[g], 1.0f);
}

// ---------------------------------------------------------------------------
extern "C" void kernel_launch(void* const* d_in, const int* in_sizes, int n_in,
                              void* d_out, int out_size, void* d_ws, size_t ws_size,
                              hipStream_t stream) {
  (void)n_in; (void)out_size; (void)ws_size;
  // Inputs per setup_inputs() order
  const float* x_in   = (const float*)d_in[0];
  const int*   eidx   = (const int*)d_in[1];
  const int*   batch  = (const int*)d_in[2];
  const float* conv_w[3] = {(const float*)d_in[3], (const float*)d_in[5], (const float*)d_in[7]};
  const float* conv_b[3] = {(const float*)d_in[4], (const float*)d_in[6], (const float*)d_in[8]};
  const float* bn_g[3]   = {(const float*)d_in[9],  (const float*)d_in[11], (const float*)d_in[13]};
  const float* bn_b[3]   = {(const float*)d_in[10], (const float*)d_in[12], (const float*)d_in[14]};
  const float* lin1_w = (const float*)d_in[15];
  const float* lin1_b = (const float*)d_in[16];
  const float* lin2_w = (const float*)d_in[17];
  const float* lin2_b = (const float*)d_in[18];
  const float* fc_w   = (const float*)d_in[19];
  const float* fc_b   = (const float*)d_in[20];

  const int N = in_sizes[2];          // nodes (batch array length) = 50000
  const int E = in_sizes[1] / 2;      // edges
  const int* src = eidx;              // edge_index[0,:]
  const int* dst = eidx + E;          // edge_index[1,:]

  // Workspace carve-out (256B aligned)
  size_t off = 0;
  auto wsAlloc = [&](size_t bytes) {
    void* p = (char*)d_ws + off;
    off += (bytes + 255) & ~(size_t)255;
    return p;
  };
  float* h      = (float*)wsAlloc((size_t)N * HDIM * 4);
  float* fA     = (float*)wsAlloc((size_t)N * HDIM * 4);
  float* fB     = (float*)wsAlloc((size_t)N * HDIM * 4);
  float* dis    = (float*)wsAlloc((size_t)N * 4);
  float* colsum = (float*)wsAlloc(HDIM * 4);
  float* colsq  = (float*)wsAlloc(HDIM * 4);
  float* scale  = (float*)wsAlloc(HDIM * 4);
  float* shift  = (float*)wsAlloc(HDIM * 4);
  float* pooled = (float*)wsAlloc((size_t)GNUM * HDIM * 4);
  float* cnt    = (float*)wsAlloc(GNUM * 4);
  float* t1     = (float*)wsAlloc((size_t)GNUM * CLSD * 4);
  float* t2     = (float*)wsAlloc((size_t)GNUM * OUTD * 4);

  // ---- normalization coefficients ----
  hipMemsetAsync(dis, 0, (size_t)N * 4, stream);
  k_deg<<<2048, 256, 0, stream>>>(dst, dis, E);
  k_dis<<<(N + 255) / 256, 256, 0, stream>>>(dis, N);

  // ---- 3 GCN layers ----
  const float* xcur = x_in;
  float* fbuf[2] = {fA, fB};
  for (int L = 0; L < 3; ++L) {
    float* agg = fbuf[L & 1];
    // h = xcur @ W   (50000 = 80 * 625 -> RT=5, grid (625, 2))
    k_gemm_wmma<5><<<dim3(N / 80, HDIM / 64), 128, 0, stream>>>(
        xcur, conv_w[L], nullptr, h, N, HDIM, HDIM, 0);
    // scatter-add messages (1 wave per edge, b128 loads)
    hipMemsetAsync(agg, 0, (size_t)N * HDIM * 4, stream);
    k_scatter<<<(E + 7) / 8, 256, 0, stream>>>(
        (const float4*)h, src, dst, dis, agg, E);
    // self-loop + bias + BN stats (b128 path)
    hipMemsetAsync(colsum, 0, HDIM * 4, stream);
    hipMemsetAsync(colsq, 0, HDIM * 4, stream);
    k_finalize<<<1024, 128, 0, stream>>>(
        (float4*)agg, (const float4*)h, dis, conv_b[L], colsum, colsq, N);
    k_bnstats<<<1, HDIM, 0, stream>>>(colsum, colsq, bn_g[L], bn_b[L], scale, shift, N);
    k_bnrelu<<<2048, 256, 0, stream>>>((float4*)agg, scale, shift, N * HDIM / 4);
    xcur = agg;
  }

  // ---- global mean pool ----
  hipMemsetAsync(pooled, 0, (size_t)GNUM * HDIM * 4, stream);
  hipMemsetAsync(cnt, 0, GNUM * 4, stream);
  k_pool<<<1024, HDIM, 0, stream>>>(xcur, batch, pooled, cnt, N);
  k_pooldiv<<<GNUM, HDIM, 0, stream>>>(pooled, cnt);

  // ---- MLP head (all WMMA) ----
  k_gemm_wmma<1><<<dim3(GNUM / 16, CLSD / 64), 128, 0, stream>>>(
      pooled, lin1_w, lin1_b, t1, GNUM, HDIM, CLSD, 0);
  k_gemm_wmma<1><<<dim3(GNUM / 16, 1), 128, 0, stream>>>(
      t1, lin2_w, lin2_b, t2, GNUM, CLSD, OUTD, 0);
  k_gemm_wmma<1><<<dim3(GNUM / 16, 1), 128, 0, stream>>>(
      t2, fc_w, fc_b, (float*)d_out, GNUM, OUTD, OUTD, 1);
}